// hyperNet_47545287966770
// MI455X (gfx1250) — compile-verified
//
#include <hip/hip_runtime.h>
#include <math.h>
#include <stdint.h>

typedef __attribute__((ext_vector_type(2))) float v2f;
typedef __attribute__((ext_vector_type(4))) float v4f;
typedef __attribute__((ext_vector_type(8))) float v8f;
typedef __attribute__((ext_vector_type(4))) int   v4i;

#define WMMA_F32(a, b, c) \
  __builtin_amdgcn_wmma_f32_16x16x4_f32(false, (a), false, (b), (short)0, (c), false, false)

constexpr int Bt  = 8;     // batch
constexpr int Nn  = 1024;  // sequence length
constexpr int CI  = 40;    // CH_IN
constexpr int CH  = 64;    // CH
constexpr int CHC = 128;   // CH_CLOSE
constexpr int CO  = 9;     // CH_OUT
constexpr int KS  = 9;     // conv kernel
constexpr int KCH = 65;    // graph channels (64 + pos)
constexpr int KA  = 68;    // padded to multiple of 4
constexpr int KC  = 32;    // K-chunk staged in LDS
constexpr int KCP = 36;    // padded LDS row stride (bank-conflict-free, 16B aligned)
constexpr float Hh = 0.1f;

// ---------- async global->LDS staging (CDNA5 path, guarded) ----------
#if __has_builtin(__builtin_amdgcn_global_load_async_to_lds_b128) && \
    __has_builtin(__builtin_amdgcn_s_wait_asynccnt)
#define HAVE_ASYNC_LDS 1
typedef __attribute__((address_space(1))) v4i* gv4ip;  // global int32x4*
typedef __attribute__((address_space(3))) v4i* sv4ip;  // LDS int32x4*
__device__ inline void lds_cp16(float* dst, const float* src) {
  gv4ip g = (gv4ip)(uintptr_t)src;  // identity: generic global == AS1
  sv4ip s = (sv4ip)(uintptr_t)dst;  // low 32 bits of generic shared == LDS offset
  __builtin_amdgcn_global_load_async_to_lds_b128(g, s, 0, 0);
}
__device__ inline void lds_join() { __builtin_amdgcn_s_wait_asynccnt(0); }
#else
__device__ inline void lds_cp16(float* dst, const float* src) {
  *(v4f*)dst = *(const v4f*)src;
}
__device__ inline void lds_join() {}
#endif

// ---------------- small helpers ----------------

__device__ inline float blk_sum(float v, float* red) {
  int t = threadIdx.x;
  red[t] = v;
  __syncthreads();
  for (int s = 128; s > 0; s >>= 1) {
    if (t < s) red[t] += red[t + s];
    __syncthreads();
  }
  float r = red[0];
  __syncthreads();
  return r;
}

// ---------------- open projection: Zb = (Kopen @ Z) * mask ----------------
// grid (Nn/16, CH/16, Bt), block 32 (one wave)
__global__ void k_open(const float* __restrict__ Zin, const float* __restrict__ Kop,
                       const float* __restrict__ mask, float* __restrict__ Zb) {
  int b = blockIdx.z, m0 = blockIdx.y * 16, n0 = blockIdx.x * 16;
  int lane = threadIdx.x & 31, half = lane >> 4, l16 = lane & 15;
  const float* Zr = Zin + (size_t)b * CI * Nn;
  v8f acc = {};
  for (int kk = 0; kk < CI; kk += 4) {
    int ka = kk + 2 * half;
    v2f a, bv;
    a.x  = Kop[(m0 + l16) * CI + ka];
    a.y  = Kop[(m0 + l16) * CI + ka + 1];
    bv.x = Zr[(size_t)ka * Nn + n0 + l16];
    bv.y = Zr[(size_t)(ka + 1) * Nn + n0 + l16];
    acc = WMMA_F32(a, bv, acc);
  }
  int col = n0 + l16;
  float mv = mask[b * Nn + col];
  for (int r = 0; r < 8; ++r) {
    int row = m0 + r + 8 * half;
    Zb[((size_t)b * CH + row) * Nn + col] = acc[r] * mv;
  }
}

// ---------------- graph prep: build centered Ya (Bt,KA,Nn) ----------------
// grid (KA, Bt), block 256
__global__ void k_prep(const float* __restrict__ Zb, const float* __restrict__ mask,
                       float* __restrict__ Ya) {
  __shared__ float red[256];
  int c = blockIdx.x, b = blockIdx.y, t = threadIdx.x;
  float* out = Ya + ((size_t)b * KA + c) * Nn;
  if (c >= KCH) {  // zero pad channels
    for (int i = 0; i < 4; ++i) out[t + 256 * i] = 0.0f;
    return;
  }
  const float* mr = mask + b * Nn;
  float x1[4];
  if (c < CH) {
    const float* xr = Zb + ((size_t)b * CH + c) * Nn;
    float s1 = 0.f, sm = 0.f;
    for (int i = 0; i < 4; ++i) {
      int n = t + 256 * i;
      s1 += xr[n] * mr[n];
      sm += mr[n];
    }
    s1 = blk_sum(s1, red);
    sm = blk_sum(sm, red);
    float mean = s1 / sm;
    float s2 = 0.f;
    for (int i = 0; i < 4; ++i) {
      int n = t + 256 * i;
      float xc = xr[n] - mean * mr[n];
      x1[i] = xc;
      s2 += xc * xc;
    }
    s2 = blk_sum(s2, red);
    float inv = rsqrtf(s2 / sm + 1e-4f);
    for (int i = 0; i < 4; ++i) x1[i] *= inv;
  } else {  // positional channel
    for (int i = 0; i < 4; ++i) {
      int n = t + 256 * i;
      x1[i] = 0.5f * ((float)n / (float)(Nn - 1)) * mr[n];
    }
  }
  // tr2dist re-centering (plain mean over N)
  float s = x1[0] + x1[1] + x1[2] + x1[3];
  s = blk_sum(s, red) / (float)Nn;
  for (int i = 0; i < 4; ++i) out[t + 256 * i] = x1[i] - s;
}

// ---------------- n2[b,n] = sum_c Ya^2 ----------------  grid (Bt), block 256
__global__ void k_n2(const float* __restrict__ Ya, float* __restrict__ n2) {
  int b = blockIdx.x, t = threadIdx.x;
  for (int i = 0; i < 4; ++i) {
    int n = t + 256 * i;
    float s = 0.f;
    for (int c = 0; c < KCH; ++c) {
      float v = Ya[((size_t)b * KA + c) * Nn + n];
      s += v * v;
    }
    n2[b * Nn + n] = s;
  }
}

// ---------------- W = exp(-dist)*mm via WMMA Gram ----------------
// grid (Nn/64, Nn/16, Bt), block 128 (4 waves, each one 16x16 tile)
__global__ void k_W(const float* __restrict__ Ya, const float* __restrict__ n2,
                    const float* __restrict__ mask, float* __restrict__ Wb) {
  int b = blockIdx.z;
  int i0 = blockIdx.y * 16;
  int j0 = (blockIdx.x * 4 + (threadIdx.x >> 5)) * 16;
  int lane = threadIdx.x & 31, half = lane >> 4, l16 = lane & 15;
  const float* Y = Ya + (size_t)b * KA * Nn;
  v8f acc = {};
  for (int kk = 0; kk < KA; kk += 4) {
    int ka = kk + 2 * half;
    v2f a, bv;
    a.x  = Y[(size_t)ka * Nn + i0 + l16];
    a.y  = Y[(size_t)(ka + 1) * Nn + i0 + l16];
    bv.x = Y[(size_t)ka * Nn + j0 + l16];
    bv.y = Y[(size_t)(ka + 1) * Nn + j0 + l16];
    acc = WMMA_F32(a, bv, acc);
  }
  const float* mr = mask + b * Nn;
  int j = j0 + l16;
  float mj = mr[j], nj = n2[b * Nn + j];
  for (int r = 0; r < 8; ++r) {
    int i = i0 + r + 8 * half;
    float d = n2[b * Nn + i] + nj - 2.0f * acc[r];
    d = fmaxf(d * (3.0f / (float)KCH), 0.0f);
    float wd = (d > 0.0f) ? sqrtf(d) : 0.0f;
    float mm = mr[i] * mj;
    Wb[((size_t)b * Nn + i) * Nn + j] = __expf(-wd * mm) * mm;
  }
}

// ---------------- Wsum[b,n] = sum_i W[b,i,n] ----------------  grid (4, Bt), block 256
__global__ void k_wsum(const float* __restrict__ Wb, float* __restrict__ Wsum) {
  int b = blockIdx.y;
  int n = blockIdx.x * 256 + threadIdx.x;
  float s = 0.f;
  for (int i = 0; i < Nn; ++i) s += Wb[((size_t)b * Nn + i) * Nn + n];
  Wsum[b * Nn + n] = s;
}

// ---------------- L = diag(Wsum) - W, in place ----------------  grid 32768, block 256
__global__ void k_Lip(float* __restrict__ Wb, const float* __restrict__ Wsum) {
  size_t idx = (size_t)blockIdx.x * 256 + threadIdx.x;
  int b = (int)(idx >> 20);
  int rem = (int)(idx & 1048575);
  int i = rem >> 10, j = rem & 1023;
  float v = Wb[idx];
  Wb[idx] = ((i == j) ? Wsum[b * Nn + i] : 0.0f) - v;
}

// ---------------- conv1d 64->128, K=9 as implicit-im2col WMMA GEMM ----------------
// A (weights) chunk staged in LDS once per block (shared by all 4 waves).
// grid (Nn/64, CHC/16, Bt), block 128
__global__ void k_conv(const float* __restrict__ Zb, const float* __restrict__ Wl,
                       const float* __restrict__ bias, const float* __restrict__ mask,
                       float* __restrict__ out, int applyMask) {
  __shared__ __align__(16) float As[16 * KCP];
  int b = blockIdx.z;
  int m0 = blockIdx.y * 16;
  int n0 = (blockIdx.x * 4 + (threadIdx.x >> 5)) * 16;
  int lane = threadIdx.x & 31, half = lane >> 4, l16 = lane & 15;
  int arow = threadIdx.x >> 3, acol = (threadIdx.x & 7) * 4;
  const float* Zr = Zb + (size_t)b * CH * Nn;
  const float* Aw = Wl + (size_t)m0 * (CH * KS);
  int n = n0 + l16;
  v8f acc = {};
  for (int kc = 0; kc < CH * KS; kc += KC) {
    __syncthreads();
    lds_cp16(&As[arow * KCP + acol], &Aw[(size_t)arow * (CH * KS) + kc + acol]);
    lds_join();
    __syncthreads();
    for (int kk = 0; kk < KC; kk += 4) {
      int koff = kk + 2 * half;
      int ka = kc + koff;
      v2f a = *(const v2f*)&As[l16 * KCP + koff];
      v2f bv;
      {
        int c = ka / KS, tt = ka - c * KS, p = n + tt - 4;
        bv.x = (p >= 0 && p < Nn) ? (Zr[(size_t)c * Nn + p] + bias[c]) : 0.0f;
      }
      {
        int k1 = ka + 1;
        int c = k1 / KS, tt = k1 - c * KS, p = n + tt - 4;
        bv.y = (p >= 0 && p < Nn) ? (Zr[(size_t)c * Nn + p] + bias[c]) : 0.0f;
      }
      acc = WMMA_F32(a, bv, acc);
    }
  }
  float mv = applyMask ? mask[b * Nn + n] : 1.0f;
  for (int r = 0; r < 8; ++r) {
    int row = m0 + r + 8 * half;
    out[((size_t)b * CHC + row) * Nn + n] = acc[r] * mv;
  }
}

// ---------------- transpose conv 128->64, K=9 as implicit GEMM ----------------
// Flipped weights gathered into LDS per chunk (computed layout), shared by 4 waves.
// grid (Nn/64, CH/16, Bt), block 128
__global__ void k_convT(const float* __restrict__ in, const float* __restrict__ Wl,
                        const float* __restrict__ mask, float* __restrict__ out,
                        int applyMask) {
  __shared__ __align__(16) float As[16 * KCP];
  int b = blockIdx.z;
  int m0 = blockIdx.y * 16;
  int n0 = (blockIdx.x * 4 + (threadIdx.x >> 5)) * 16;
  int lane = threadIdx.x & 31, half = lane >> 4, l16 = lane & 15;
  int arow = threadIdx.x >> 3, acol = (threadIdx.x & 7) * 4;
  const float* Xr = in + (size_t)b * CHC * Nn;
  int n = n0 + l16;
  int m = m0 + arow;  // channel index this thread stages weights for
  v8f acc = {};
  for (int kc = 0; kc < CHC * KS; kc += KC) {
    __syncthreads();
    {
      v4f w;
      for (int j = 0; j < 4; ++j) {
        int k = kc + acol + j;
        int o = k / KS, tt = k - o * KS;
        w[j] = Wl[(size_t)o * (CH * KS) + m * KS + (KS - 1 - tt)];
      }
      *(v4f*)&As[arow * KCP + acol] = w;
    }
    __syncthreads();
    for (int kk = 0; kk < KC; kk += 4) {
      int koff = kk + 2 * half;
      int ka = kc + koff;
      v2f a = *(const v2f*)&As[l16 * KCP + koff];
      v2f bv;
      {
        int o = ka / KS, tt = ka - o * KS, p = n + tt - 4;
        bv.x = (p >= 0 && p < Nn) ? Xr[(size_t)o * Nn + p] : 0.0f;
      }
      {
        int k1 = ka + 1;
        int o = k1 / KS, tt = k1 - o * KS, p = n + tt - 4;
        bv.y = (p >= 0 && p < Nn) ? Xr[(size_t)o * Nn + p] : 0.0f;
      }
      acc = WMMA_F32(a, bv, acc);
    }
  }
  float mv = applyMask ? mask[b * Nn + n] : 1.0f;
  for (int r = 0; r < 8; ++r) {
    int row = m0 + r + 8 * half;
    out[((size_t)b * CH + row) * Nn + n] = acc[r] * mv;
  }
}

// ---------------- out = (A @ L) * mask ; A is (Bt, M, Nn), L is (Bt, Nn, Nn) ----------------
// A chunk staged in LDS via async copy, shared by 4 waves; L read coalesced from L2.
// grid (Nn/64, M/16, Bt), block 128
__global__ void k_matL(const float* __restrict__ A, const float* __restrict__ L,
                       const float* __restrict__ mask, float* __restrict__ out,
                       int M, int applyMask) {
  __shared__ __align__(16) float As[16 * KCP];
  int b = blockIdx.z;
  int m0 = blockIdx.y * 16;
  int n0 = (blockIdx.x * 4 + (threadIdx.x >> 5)) * 16;
  int lane = threadIdx.x & 31, half = lane >> 4, l16 = lane & 15;
  int arow = threadIdx.x >> 3, acol = (threadIdx.x & 7) * 4;
  const float* Ar = A + (size_t)b * M * Nn + (size_t)m0 * Nn;
  const float* Lr = L + (size_t)b * Nn * Nn;
  int n = n0 + l16;
  v8f acc = {};
  for (int kc = 0; kc < Nn; kc += KC) {
    __syncthreads();
    lds_cp16(&As[arow * KCP + acol], &Ar[(size_t)arow * Nn + kc + acol]);
    lds_join();
    __syncthreads();
    for (int kk = 0; kk < KC; kk += 4) {
      int koff = kk + 2 * half;
      int ka = kc + koff;
      if (kk == 0 && ka + KC < Nn)  // stage upcoming L rows into cache
        __builtin_prefetch(&Lr[(size_t)(ka + KC) * Nn + n], 0, 1);
      v2f a = *(const v2f*)&As[l16 * KCP + koff];
      v2f bv;
      bv.x = Lr[(size_t)ka * Nn + n];
      bv.y = Lr[(size_t)(ka + 1) * Nn + n];
      acc = WMMA_F32(a, bv, acc);
    }
  }
  float mv = applyMask ? mask[b * Nn + n] : 1.0f;
  for (int r = 0; r < 8; ++r) {
    int row = m0 + r + 8 * half;
    out[((size_t)b * M + row) * Nn + n] = acc[r] * mv;
  }
}

// ---------------- masked spatial mean per (b,ch) ----------------  grid (CHC, Bt), block 256
__global__ void k_rowmean(const float* __restrict__ T, const float* __restrict__ mask,
                          float* __restrict__ means) {
  __shared__ float red[256];
  int c = blockIdx.x, b = blockIdx.y, t = threadIdx.x;
  const float* xr = T + ((size_t)b * CHC + c) * Nn;
  const float* mr = mask + b * Nn;
  float s = 0.f, sm = 0.f;
  for (int i = 0; i < 4; ++i) {
    int n = t + 256 * i;
    s += xr[n] * mr[n];
    sm += mr[n];
  }
  s = blk_sum(s, red);
  sm = blk_sum(sm, red);
  if (t == 0) means[b * CHC + c] = s / sm;
}

// ---------------- per-(b,n): subtract mean, mask, channel-normalize, relu (in place) ----------------
// grid 32, block 256
__global__ void k_normrelu(float* __restrict__ T, const float* __restrict__ mask,
                           const float* __restrict__ means) {
  int g = blockIdx.x * 256 + threadIdx.x;
  int b = g >> 10, n = g & 1023;
  float mv = mask[b * Nn + n];
  float s = 0.f;
  for (int c = 0; c < CHC; ++c) {
    float v = (T[((size_t)b * CHC + c) * Nn + n] - means[b * CHC + c]) * mv;
    s += v * v;
  }
  float inv = rsqrtf(s + 0.001f);
  for (int c = 0; c < CHC; ++c) {
    float v = (T[((size_t)b * CHC + c) * Nn + n] - means[b * CHC + c]) * mv;
    T[((size_t)b * CHC + c) * Nn + n] = fmaxf(v * inv, 0.0f);
  }
}

// ---------------- Z -= H*(U0+U1) ----------------  grid 2048, block 256
__global__ void k_update(float* __restrict__ Zb, const float* __restrict__ U0,
                         const float* __restrict__ U1) {
  size_t i = (size_t)blockIdx.x * 256 + threadIdx.x;
  Zb[i] -= Hh * (U0[i] + U1[i]);
}

// ---------------- close projection: Zo = (Kclose @ Z)*mask ----------------
// grid (Nn/16, 1, Bt), block 32
__global__ void k_close(const float* __restrict__ Zb, const float* __restrict__ Kcl,
                        const float* __restrict__ mask, float* __restrict__ Zo) {
  int b = blockIdx.z, n0 = blockIdx.x * 16;
  int lane = threadIdx.x & 31, half = lane >> 4, l16 = lane & 15;
  v8f acc = {};
  for (int kk = 0; kk < CH; kk += 4) {
    int ka = kk + 2 * half;
    v2f a, bv;
    a.x  = (l16 < CO) ? Kcl[l16 * CH + ka] : 0.0f;
    a.y  = (l16 < CO) ? Kcl[l16 * CH + ka + 1] : 0.0f;
    bv.x = Zb[((size_t)b * CH + ka) * Nn + n0 + l16];
    bv.y = Zb[((size_t)b * CH + ka + 1) * Nn + n0 + l16];
    acc = WMMA_F32(a, bv, acc);
  }
  int col = n0 + l16;
  float mv = mask[b * Nn + col];
  for (int r = 0; r < 8; ++r) {
    int row = r + 8 * half;
    if (row < CO) Zo[((size_t)b * CO + row) * Nn + col] = acc[r] * mv;
  }
}

// ---------------- prep for output distances: center 3-ch groups, n2 ----------------
// grid (3, Bt), block 256
__global__ void k_dprep(const float* __restrict__ Zo, float* __restrict__ Yd,
                        float* __restrict__ n2d) {
  __shared__ float red[256];
  int g = blockIdx.x, b = blockIdx.y, t = threadIdx.x;
  float nn[4] = {0.f, 0.f, 0.f, 0.f};
  for (int c = 0; c < 3; ++c) {
    int ch = g * 3 + c;
    const float* xr = Zo + ((size_t)b * CO + ch) * Nn;
    float s = 0.f;
    for (int i = 0; i < 4; ++i) s += xr[t + 256 * i];
    s = blk_sum(s, red) / (float)Nn;
    for (int i = 0; i < 4; ++i) {
      int n = t + 256 * i;
      float v = xr[n] - s;
      Yd[((size_t)b * CO + ch) * Nn + n] = v;
      nn[i] += v * v;
    }
  }
  for (int i = 0; i < 4; ++i) n2d[((size_t)b * 3 + g) * Nn + t + 256 * i] = nn[i];
}

// ---------------- output distance matrices (K=3, VALU) ----------------
// grid (Nn/16, Nn/16, Bt*3), block 256
__global__ void k_dist(const float* __restrict__ Yd, const float* __restrict__ n2d,
                       float* __restrict__ out) {
  int bz = blockIdx.z, b = bz / 3, g = bz % 3;
  int ti = threadIdx.x >> 4, tj = threadIdx.x & 15;
  int i = blockIdx.y * 16 + ti, j = blockIdx.x * 16 + tj;
  const float* Y = Yd + ((size_t)b * CO + g * 3) * Nn;
  float dot = Y[i] * Y[j] + Y[Nn + i] * Y[Nn + j] + Y[2 * Nn + i] * Y[2 * Nn + j];
  float d = n2d[((size_t)b * 3 + g) * Nn + i] + n2d[((size_t)b * 3 + g) * Nn + j] - 2.0f * dot;
  d = fmaxf(d, 0.0f);  // 3*d/k with k=3 => d
  out[(((size_t)b * 3 + g) * Nn + i) * Nn + j] = (d > 0.0f) ? sqrtf(d) : 0.0f;
}

// ---------------- host ----------------
extern "C" void kernel_launch(void* const* d_in, const int* in_sizes, int n_in,
                              void* d_out, int out_size, void* d_ws, size_t ws_size,
                              hipStream_t stream) {
  const float* Zin  = (const float*)d_in[0];
  const float* mask = (const float*)d_in[1];
  const float* Kop  = (const float*)d_in[2];
  const float* Kcl  = (const float*)d_in[3];
  const float* Wk   = (const float*)d_in[4];
  const float* Bias = (const float*)d_in[5];
  float* out = (float*)d_out;

  float* ws    = (float*)d_ws;
  float* Zb    = ws;                                // 8*64*1024
  float* Ya    = Zb + (size_t)Bt * CH * Nn;         // 8*68*1024
  float* n2    = Ya + (size_t)Bt * KA * Nn;         // 8*1024
  float* Wsum  = n2 + (size_t)Bt * Nn;              // 8*1024
  float* Lb    = Wsum + (size_t)Bt * Nn;            // 8*1024*1024 (W, then L in place)
  float* T0    = Lb + (size_t)Bt * Nn * Nn;         // 8*128*1024
  float* T1    = T0 + (size_t)Bt * CHC * Nn;        // 8*128*1024
  float* T2    = T1 + (size_t)Bt * CHC * Nn;        // 8*128*1024
  float* U0    = T2 + (size_t)Bt * CHC * Nn;        // 8*64*1024
  float* U1    = U0 + (size_t)Bt * CH * Nn;         // 8*64*1024
  float* means = U1 + (size_t)Bt * CH * Nn;         // 8*128
  float* Yd    = means + Bt * CHC;                  // 8*9*1024
  float* n2d   = Yd + (size_t)Bt * CO * Nn;         // 8*3*1024

  float* Zo = out + (size_t)Bt * 3 * Nn * Nn;  // final Z lives in d_out tail

  // 1. open projection
  k_open<<<dim3(Nn / 16, CH / 16, Bt), 32, 0, stream>>>(Zin, Kop, mask, Zb);

  // 2. graph Laplacian (reference recomputes the identical L at i==0; do it once)
  k_prep<<<dim3(KA, Bt), 256, 0, stream>>>(Zb, mask, Ya);
  k_n2<<<dim3(Bt), 256, 0, stream>>>(Ya, n2);
  k_W<<<dim3(Nn / 64, Nn / 16, Bt), 128, 0, stream>>>(Ya, n2, mask, Lb);
  k_wsum<<<dim3(4, Bt), 256, 0, stream>>>(Lb, Wsum);
  k_Lip<<<dim3((Bt * Nn * Nn) / 256), 256, 0, stream>>>(Lb, Wsum);

  // 3. layers
  for (int l = 0; l < 10; ++l) {
    const float* W0 = Wk + (size_t)(l * 2 + 0) * CHC * CH * KS;
    const float* W1 = Wk + (size_t)(l * 2 + 1) * CHC * CH * KS;
    const float* B0 = Bias + (size_t)(l * 2 + 0) * CH;
    const float* B1 = Bias + (size_t)(l * 2 + 1) * CH;

    k_conv<<<dim3(Nn / 64, CHC / 16, Bt), 128, 0, stream>>>(Zb, W0, B0, mask, T0, 1);
    k_conv<<<dim3(Nn / 64, CHC / 16, Bt), 128, 0, stream>>>(Zb, W1, B1, mask, T1, 0);
    k_matL<<<dim3(Nn / 64, CHC / 16, Bt), 128, 0, stream>>>(T1, Lb, mask, T2, CHC, 1);

    k_rowmean<<<dim3(CHC, Bt), 256, 0, stream>>>(T0, mask, means);
    k_normrelu<<<dim3((Bt * Nn) / 256), 256, 0, stream>>>(T0, mask, means);
    k_rowmean<<<dim3(CHC, Bt), 256, 0, stream>>>(T2, mask, means);
    k_normrelu<<<dim3((Bt * Nn) / 256), 256, 0, stream>>>(T2, mask, means);

    k_convT<<<dim3(Nn / 64, CH / 16, Bt), 128, 0, stream>>>(T0, W0, mask, U0, 1);
    k_convT<<<dim3(Nn / 64, CH / 16, Bt), 128, 0, stream>>>(T2, W1, mask, T1, 0);
    k_matL<<<dim3(Nn / 64, CH / 16, Bt), 128, 0, stream>>>(T1, Lb, mask, U1, CH, 1);

    k_update<<<dim3((Bt * CH * Nn) / 256), 256, 0, stream>>>(Zb, U0, U1);
  }

  // 4. close projection + output distance matrices
  k_close<<<dim3(Nn / 16, 1, Bt), 32, 0, stream>>>(Zb, Kcl, mask, Zo);
  k_dprep<<<dim3(3, Bt), 256, 0, stream>>>(Zo, Yd, n2d);
  k_dist<<<dim3(Nn / 16, Nn / 16, Bt * 3), 256, 0, stream>>>(Yd, n2d, out);
}